// OrthogonalWight_83519934038257
// MI455X (gfx1250) — compile-verified
//
#include <hip/hip_runtime.h>
#include <math.h>

// ---------------------------------------------------------------------------
// Blocked Householder (compact WY) on MI455X / gfx1250, fp32 WMMA +
// double-buffered GLOBAL_LOAD_ASYNC_TO_LDS_B128 staging.
//   Q = I; for each 128-column block b of normalized U then V:
//       G_b = Y_b^T Y_b            (128x128, K=4096 GEMM)     <1,0,set>
//       S_b : s_j = 2 e_j - S g_j  (tiny per-block recurrence)
//       W_b = Y_b S_b              (4096x128, K=128 GEMM)     <0,0,set>
//       T   = Q W_b                (4096x128, K=4096 GEMM)    <0,0,set>
//       Q  -= T Y_b^T              (4096x4096, K=128 GEMM)    <0,1,sub>
//   Result Q == Qu @ Qv lands directly in d_out (reshape is a no-op).
// ---------------------------------------------------------------------------

typedef float v2f __attribute__((ext_vector_type(2)));
typedef float v8f __attribute__((ext_vector_type(8)));
typedef int   v4i __attribute__((vector_size(16)));   // builtin's pointee type

#define N_DIM 4096
#define NB    128
#define NBLK  32
#define BM    128
#define BN    128
#define BK    32
#define SK    (BK + 4)   // 36 floats: k-contiguous row stride (16B aligned)
#define SM    (BM + 4)   // 132 floats: m/n-contiguous row stride (16B aligned)

// one 16-byte async copy: global -> LDS, ASYNCcnt-tracked
__device__ __forceinline__ void async_cp16(const float* gsrc, float* ldst)
{
    __builtin_amdgcn_global_load_async_to_lds_b128(
        (v4i*)gsrc, (v4i*)ldst, 0, 0);
}

// ---------------------------------------------------------------------------
// Generic tiled GEMM:  C (+)= op(A) * op(B), tile 128x128x32, 256 thr = 8 waves.
// TA: A[m][k] = A[k*lda + m] (else A[m*lda + k])
// TB: B[k][n] = B[n*ldb + k] (else B[k*ldb + n])
// MODE 0: C = A*B ; MODE 1: C -= A*B
// LDS layout follows global-contiguous axis so ASYNC_B128 writes are legal:
//   TA=0: As[m][SK] (k-contig)   TA=1: As[k][SM] (m-contig)
//   TB=0: Bs[k][SM] (n-contig)   TB=1: Bs[n][SK] (k-contig)
// All dims multiples of 128 (M,N) and 32 (K): no bounds checks.
// ---------------------------------------------------------------------------
template<int TA, int TB, int MODE>
__global__ __launch_bounds__(256) void gemm_wmma_f32(
    const float* __restrict__ A, int lda,
    const float* __restrict__ B, int ldb,
    float* __restrict__ C, int ldc, int K)
{
    constexpr int AELEM = TA ? (BK * SM) : (BM * SK);
    constexpr int BELEM = TB ? (BN * SK) : (BK * SM);
    __shared__ float As[2][AELEM];
    __shared__ float Bs[2][BELEM];

    const int tid  = threadIdx.x;
    const int lane = tid & 31;
    const int wv   = tid >> 5;          // wave 0..7
    const int ln   = lane & 15;
    const int kh   = lane >> 4;         // 0/1: K-pair (A,B) / M-half (C,D)
    const int m0   = blockIdx.x * BM;
    const int n0   = blockIdx.y * BN;
    const int m0w  = wv * 16;           // wave's 16-row strip

    // ---- async stage of one 32-deep K chunk (8 x b128 per thread) -------
    auto stage = [&](int k0, int buf) {
        if (TA == 0) {                      // A k-contiguous -> As[m][SK]
            const int kq = (tid & 7) * 4;
            const int mb = tid >> 3;
            #pragma unroll
            for (int it = 0; it < 4; ++it) {
                const int m = mb + it * 32;
                async_cp16(A + (size_t)(m0 + m) * lda + (k0 + kq),
                           &As[buf][m * SK + kq]);
            }
        } else {                            // A m-contiguous -> As[k][SM]
            const int mq = (tid & 31) * 4;
            const int kb = tid >> 5;
            #pragma unroll
            for (int it = 0; it < 4; ++it) {
                const int k = kb + it * 8;
                async_cp16(A + (size_t)(k0 + k) * lda + (m0 + mq),
                           &As[buf][k * SM + mq]);
            }
        }
        if (TB == 0) {                      // B n-contiguous -> Bs[k][SM]
            const int nq = (tid & 31) * 4;
            const int kb = tid >> 5;
            #pragma unroll
            for (int it = 0; it < 4; ++it) {
                const int k = kb + it * 8;
                async_cp16(B + (size_t)(k0 + k) * ldb + (n0 + nq),
                           &Bs[buf][k * SM + nq]);
            }
        } else {                            // B k-contiguous -> Bs[n][SK]
            const int kq = (tid & 7) * 4;
            const int nb = tid >> 3;
            #pragma unroll
            for (int it = 0; it < 4; ++it) {
                const int n = nb + it * 32;
                async_cp16(B + (size_t)(n0 + n) * ldb + (k0 + kq),
                           &Bs[buf][n * SK + kq]);
            }
        }
    };

    v8f acc[8];
    #pragma unroll
    for (int t = 0; t < 8; ++t)
        #pragma unroll
        for (int c = 0; c < 8; ++c) acc[t][c] = 0.0f;

    const int nch = K / BK;
    stage(0, 0);

    for (int i = 0; i < nch; ++i) {
        const int cur = i & 1;
        const bool more = (i + 1) < nch;
        if (more) stage((i + 1) * BK, cur ^ 1);

        // own async ops: chunk i is resident once count drops to the 8 just
        // issued for chunk i+1 (in-order completion); barrier makes all
        // waves' writes visible.
        if (more) __builtin_amdgcn_s_wait_asynccnt(8);
        else      __builtin_amdgcn_s_wait_asynccnt(0);
        __syncthreads();

        // ---- 8 x (K=4) WMMA steps over the chunk ------------------------
        #pragma unroll
        for (int kk = 0; kk < BK; kk += 4) {
            v2f a;  // A frag: lane row = m0w+ln, k = kk + 2*kh + {0,1}
            if (TA == 0) {
                a = *(const v2f*)&As[cur][(m0w + ln) * SK + kk + 2 * kh];
            } else {
                a.x = As[cur][(kk + 2 * kh + 0) * SM + m0w + ln];
                a.y = As[cur][(kk + 2 * kh + 1) * SM + m0w + ln];
            }
            #pragma unroll
            for (int nt = 0; nt < 8; ++nt) {
                v2f b;  // B frag: col = nt*16+ln, same k pair
                if (TB == 0) {
                    b.x = Bs[cur][(kk + 2 * kh + 0) * SM + nt * 16 + ln];
                    b.y = Bs[cur][(kk + 2 * kh + 1) * SM + nt * 16 + ln];
                } else {
                    b = *(const v2f*)&Bs[cur][(nt * 16 + ln) * SK + kk + 2 * kh];
                }
                acc[nt] = __builtin_amdgcn_wmma_f32_16x16x4_f32(
                    false, a, false, b, (short)0, acc[nt], false, false);
            }
        }
        __syncthreads();   // done reading buf `cur` before it is re-staged
    }

    // ---- epilogue: D layout -> VGPR i holds row (8*kh + i), col ln ------
    #pragma unroll
    for (int nt = 0; nt < 8; ++nt) {
        #pragma unroll
        for (int i = 0; i < 8; ++i) {
            const int row = m0 + m0w + kh * 8 + i;
            const int col = n0 + nt * 16 + ln;
            const size_t idx = (size_t)row * ldc + col;
            const float v = acc[nt][i];
            if (MODE) C[idx] -= v;
            else      C[idx]  = v;
        }
    }
}

// ---------------------------------------------------------------------------
// Column 2-norms: one thread per column (coalesced across columns).
// ---------------------------------------------------------------------------
__global__ void col_norms_kernel(const float* __restrict__ X, float* __restrict__ norms)
{
    const int c = blockIdx.x * blockDim.x + threadIdx.x;
    float s = 0.0f;
    for (int r = 0; r < N_DIM; ++r) {
        const float v = X[(size_t)r * N_DIM + c];
        s += v * v;
    }
    norms[c] = sqrtf(s);
}

__global__ void normalize_kernel(const float* __restrict__ X,
                                 const float* __restrict__ norms,
                                 float* __restrict__ Y)
{
    const size_t idx = (size_t)blockIdx.x * blockDim.x + threadIdx.x;
    const int col = (int)(idx & (N_DIM - 1));
    Y[idx] = X[idx] / norms[col];
}

__global__ void identity_kernel(float* __restrict__ Q)
{
    const size_t idx = (size_t)blockIdx.x * blockDim.x + threadIdx.x;
    const int row = (int)(idx >> 12);          // N_DIM == 2^12
    const int col = (int)(idx & (N_DIM - 1));
    Q[idx] = (row == col) ? 1.0f : 0.0f;
}

// ---------------------------------------------------------------------------
// Per-block WY coefficient recurrence: s_j = 2 e_j - S_{j-1} g_j,
// g_j[i] = G_b[i][j]. One 128-thread workgroup per block; thread r owns row r.
// ---------------------------------------------------------------------------
__global__ __launch_bounds__(128) void panel_kernel(const float* __restrict__ G,
                                                    float* __restrict__ S)
{
    const int b = blockIdx.x;
    const int r = threadIdx.x;
    const float* Gb = G + (size_t)b * NB * NB;
    float*       Sb = S + (size_t)b * NB * NB;

    float srow[NB];                 // row r of S (private)
    __shared__ float gcol[NB];

    for (int j = 0; j < NB; ++j) {
        gcol[r] = Gb[(size_t)r * NB + j];   // g_j[i] = G[i][j]
        __syncthreads();
        float acc = (r == j) ? 2.0f : 0.0f;
        for (int i = 0; i < j; ++i) acc -= srow[i] * gcol[i];
        srow[j] = acc;
        __syncthreads();
    }
    for (int i = 0; i < NB; ++i) Sb[(size_t)r * NB + i] = srow[i];
}

// ---------------------------------------------------------------------------
// Host-side orchestration (all launches on `stream`; graph-capture safe).
// Workspace layout (floats):
//   Y    : N*N     Wall : N*N     G,S : 32*128*128 each
//   T    : N*128   norms: N       (~141 MB total)
// ---------------------------------------------------------------------------
extern "C" void kernel_launch(void* const* d_in, const int* in_sizes, int n_in,
                              void* d_out, int out_size, void* d_ws, size_t ws_size,
                              hipStream_t stream)
{
    const float* Umat = (const float*)d_in[0];
    const float* Vmat = (const float*)d_in[1];
    float* Q  = (float*)d_out;
    float* ws = (float*)d_ws;

    const size_t NN = (size_t)N_DIM * N_DIM;
    float* Y     = ws;
    float* Wall  = ws + NN;
    float* G     = ws + 2 * NN;
    float* S     = G + (size_t)NBLK * NB * NB;
    float* T     = S + (size_t)NBLK * NB * NB;
    float* norms = T + (size_t)N_DIM * NB;

    // Q = I
    identity_kernel<<<dim3((unsigned)(NN / 256)), 256, 0, stream>>>(Q);

    const float* Xs[2] = { Umat, Vmat };
    for (int x = 0; x < 2; ++x) {
        const float* X = Xs[x];

        // normalize columns
        col_norms_kernel<<<dim3(N_DIM / 256), 256, 0, stream>>>(X, norms);
        normalize_kernel<<<dim3((unsigned)(NN / 256)), 256, 0, stream>>>(X, norms, Y);

        // G_b = Y_b^T Y_b   (M=128, N=128, K=4096; A transposed)
        for (int b = 0; b < NBLK; ++b) {
            gemm_wmma_f32<1, 0, 0><<<dim3(1, 1), 256, 0, stream>>>(
                Y + (size_t)b * NB, N_DIM,
                Y + (size_t)b * NB, N_DIM,
                G + (size_t)b * NB * NB, NB, N_DIM);
        }

        // S_b (all 32 blocks in one launch)
        panel_kernel<<<dim3(NBLK), NB, 0, stream>>>(G, S);

        // W_b = Y_b @ S_b   (M=4096, N=128, K=128)
        for (int b = 0; b < NBLK; ++b) {
            gemm_wmma_f32<0, 0, 0><<<dim3(N_DIM / BM, 1), 256, 0, stream>>>(
                Y + (size_t)b * NB, N_DIM,
                S + (size_t)b * NB * NB, NB,
                Wall + (size_t)b * NB, N_DIM, NB);
        }

        // sequential block applications:  Q <- Q - (Q W_b) Y_b^T
        for (int b = 0; b < NBLK; ++b) {
            // T = Q @ W_b   (M=4096, N=128, K=4096)
            gemm_wmma_f32<0, 0, 0><<<dim3(N_DIM / BM, 1), 256, 0, stream>>>(
                Q, N_DIM,
                Wall + (size_t)b * NB, N_DIM,
                T, NB, N_DIM);
            // Q -= T @ Y_b^T   (M=4096, N=4096, K=128; B transposed)
            gemm_wmma_f32<0, 1, 1><<<dim3(N_DIM / BM, N_DIM / BN), 256, 0, stream>>>(
                T, NB,
                Y + (size_t)b * NB, N_DIM,
                Q, N_DIM, NB);
        }
    }
}